// N3_63788854280767
// MI455X (gfx1250) — compile-verified
//
#include <hip/hip_runtime.h>

// CDNA5 / gfx1250, wave32, single-wave workgroups.
// Each wave owns 2 independent chains of 4 batches each (8 batches/wave).
// V_WMMA_F32_16X16X4_F32: A = 4 stacked running 4x4 prefixes (16x4),
// B = 4 side-by-side next-step matrices (4x16); diagonal 4x4 blocks of the
// 16x16 D are the new prefixes. Diagonal extraction via a wave-private LDS
// tile, relying on the ISA guarantee that same-wave LDS ops execute in order
// (no barrier, no storecnt coupling). B operands are loaded through a
// depth-16 register pipeline so global-load latency stays off the serial
// dependency chain.

typedef __attribute__((ext_vector_type(2))) float v2f;  // A/B: 2 VGPRs
typedef __attribute__((ext_vector_type(8))) float v8f;  // C/D: 8 VGPRs

#define NCHAIN 512
#define PIPE 16   // load-ahead depth (steps); NCHAIN % PIPE == 0 assumed

__device__ __forceinline__ v8f mm16x16x4(v2f A, v2f Bv) {
  v8f C = {0.f, 0.f, 0.f, 0.f, 0.f, 0.f, 0.f, 0.f};
  return __builtin_amdgcn_wmma_f32_16x16x4_f32(false, A, false, Bv, (short)0,
                                               C, false, false);
}

__global__ __launch_bounds__(32) void prefix4x4_wmma(
    const float* __restrict__ I,   // [B, NCHAIN, 4, 4]
    float* __restrict__ out,       // [B, NCHAIN+1, 4, 4]
    int nchain)
{
  __shared__ float lds[2][16][16];   // one 16x16 D tile per chain

  const int lane = (int)(threadIdx.x & 31);
  const int half = lane >> 4;               // 0: lanes 0-15, 1: lanes 16-31
  const int b0   = (int)blockIdx.x << 3;    // 8 batches: 2 chains x 4

  // A-operand (16x4): lanes 0-15 = row M=lane, K=0,1; lanes 16-31 = K=2,3.
  const int Arow = lane & 15, Ak0 = half * 2;
  const int Ablk = Arow >> 2, Ar = Arow & 3;
  const int Acol = Ablk * 4 + Ak0;          // diagonal-block column base in D
  // B-operand (4x16): lanes 0-15 = col N=lane rows K=0,1; lanes 16-31 K=2,3.
  const int Bn = lane & 15, Bj = Bn >> 2, Bc = Bn & 3, Bk0 = half * 2;

  const size_t mstride = (size_t)nchain * 16;
  const float* IbB0 = I + (size_t)(b0 + Bj) * mstride;
  const float* IbB1 = I + (size_t)(b0 + 4 + Bj) * mstride;
  const float* IbA0 = I + (size_t)(b0 + Ablk) * mstride;
  const float* IbA1 = I + (size_t)(b0 + 4 + Ablk) * mstride;
  float* outA0 = out + (size_t)(b0 + Ablk) * (size_t)(nchain + 1) * 16
                     + Ar * 4 + Ak0;
  float* outA1 = out + (size_t)(b0 + 4 + Ablk) * (size_t)(nchain + 1) * 16
                     + Ar * 4 + Ak0;

  // P_0 = I[b,0]; emit out[b,0].
  v2f A0, A1;
  A0.x = IbA0[Ar * 4 + Ak0]; A0.y = IbA0[Ar * 4 + Ak0 + 1];
  A1.x = IbA1[Ar * 4 + Ak0]; A1.y = IbA1[Ar * 4 + Ak0 + 1];
  outA0[0] = A0.x; outA0[1] = A0.y;
  outA1[0] = A1.x; outA1[1] = A1.y;

  // Fill the B register pipeline with steps k = 1 .. PIPE.
  v2f p0[PIPE], p1[PIPE];
#pragma unroll
  for (int u = 0; u < PIPE; ++u) {
    const float* m0 = IbB0 + (size_t)(1 + u) * 16;
    const float* m1 = IbB1 + (size_t)(1 + u) * 16;
    p0[u].x = m0[Bk0 * 4 + Bc]; p0[u].y = m0[(Bk0 + 1) * 4 + Bc];
    p1[u].x = m1[Bk0 * 4 + Bc]; p1[u].y = m1[(Bk0 + 1) * 4 + Bc];
  }

  int k0 = 1;
  for (; k0 + PIPE <= nchain; k0 += PIPE) {
#pragma unroll
    for (int u = 0; u < PIPE; ++u) {
      const int k = k0 + u;
      int kn = k + PIPE; kn = (kn < nchain) ? kn : (nchain - 1);
      v2f B0 = p0[u], B1 = p1[u];

      // Refill this pipeline slot for step k+PIPE; prefetch further ahead.
      const float* m0 = IbB0 + (size_t)kn * 16;
      const float* m1 = IbB1 + (size_t)kn * 16;
      __builtin_prefetch(m0 + 16 * PIPE, 0, 3);
      __builtin_prefetch(m1 + 16 * PIPE, 0, 3);
      p0[u].x = m0[Bk0 * 4 + Bc]; p0[u].y = m0[(Bk0 + 1) * 4 + Bc];
      p1[u].x = m1[Bk0 * 4 + Bc]; p1[u].y = m1[(Bk0 + 1) * 4 + Bc];

      v8f D0 = mm16x16x4(A0, B0);
      v8f D1 = mm16x16x4(A1, B1);
#pragma unroll
      for (int v = 0; v < 8; ++v) {
        lds[0][v + half * 8][lane & 15] = D0[v];
        lds[1][v + half * 8][lane & 15] = D1[v];
      }
      asm volatile("" ::: "memory");  // same-wave LDS is in-order: compiler fence only
      A0.x = lds[0][Arow][Acol]; A0.y = lds[0][Arow][Acol + 1];
      A1.x = lds[1][Arow][Acol]; A1.y = lds[1][Arow][Acol + 1];
      asm volatile("" ::: "memory");

      float* o0 = outA0 + (size_t)k * 16;
      float* o1 = outA1 + (size_t)k * 16;
      o0[0] = A0.x; o0[1] = A0.y;
      o1[0] = A1.x; o1[1] = A1.y;
    }
  }

  // Epilogue: drain remaining pipeline slots (PIPE-1 steps when NCHAIN%PIPE==0).
#pragma unroll
  for (int u = 0; u < PIPE - 1; ++u) {
    const int k = k0 + u;
    if (k < nchain) {
      v2f B0 = p0[u], B1 = p1[u];
      v8f D0 = mm16x16x4(A0, B0);
      v8f D1 = mm16x16x4(A1, B1);
#pragma unroll
      for (int v = 0; v < 8; ++v) {
        lds[0][v + half * 8][lane & 15] = D0[v];
        lds[1][v + half * 8][lane & 15] = D1[v];
      }
      asm volatile("" ::: "memory");
      A0.x = lds[0][Arow][Acol]; A0.y = lds[0][Arow][Acol + 1];
      A1.x = lds[1][Arow][Acol]; A1.y = lds[1][Arow][Acol + 1];
      asm volatile("" ::: "memory");

      float* o0 = outA0 + (size_t)k * 16;
      float* o1 = outA1 + (size_t)k * 16;
      o0[0] = A0.x; o0[1] = A0.y;
      o1[0] = A1.x; o1[1] = A1.y;
    }
  }

  // out[b, nchain] duplicates out[b, nchain-1].
  float* o0 = outA0 + (size_t)nchain * 16;
  float* o1 = outA1 + (size_t)nchain * 16;
  o0[0] = A0.x; o0[1] = A0.y;
  o1[0] = A1.x; o1[1] = A1.y;
}

extern "C" void kernel_launch(void* const* d_in, const int* in_sizes, int n_in,
                              void* d_out, int out_size, void* d_ws, size_t ws_size,
                              hipStream_t stream) {
  (void)n_in; (void)out_size; (void)d_ws; (void)ws_size;
  const float* I = (const float*)d_in[0];
  float* out = (float*)d_out;
  const int nchain = NCHAIN;
  const int nbatch = in_sizes[0] / (nchain * 16);   // 4096
  dim3 grid(nbatch / 8);   // 2 chains x 4 batches per wave, 1 wave per block
  dim3 block(32);
  hipLaunchKernelGGL(prefix4x4_wmma, grid, block, 0, stream, I, out, nchain);
}